// LoopyBeliefPropagation_21122649161878
// MI455X (gfx1250) — compile-verified
//
#include <hip/hip_runtime.h>
#include <hip/hip_bf16.h>

typedef float v2f __attribute__((ext_vector_type(2)));
typedef float v8f __attribute__((ext_vector_type(8)));

#define SDIM 128
#define NBATCH 8
#define MSTR 132               // padded LDS row stride (floats) to break bank conflicts
#define LN2F 0.6931471805599453f

// One workgroup per (i, b). 512 threads = 16 waves (wave32).
// LDS-resident m0/m1 (the [j,k] slice of m_sib for this (i,b)), softplus(s_sib)
// in registers. 3 belief-propagation iterations fully on-chip; only the final
// b vector (marginals slice) is written to global memory.
__global__ __launch_bounds__(512) void lbp_fused_kernel(
    const float* __restrict__ s_edge,          // [B,S,S,2]
    const float* __restrict__ s_sib,           // [B,S,S,S]
    const unsigned char* __restrict__ mask,    // [B,S,S] bool (1 byte)
    float* __restrict__ out)                   // [B,S,S,2]
{
    __shared__ __align__(16) float sm0[SDIM * MSTR];   // m_sib q=0 slice [j][k]
    __shared__ __align__(16) float sm1[SDIM * MSTR];   // m_sib q=1 slice [j][k]
    __shared__ float fm[SDIM];                         // fmask[k] = mask[b,k,i]
    __shared__ float pe[SDIM * 2];                     // p_edge[j][q] = s_edge[b,j,i,q]
    __shared__ float bs0[SDIM];                        // sum_k m0[j][k]*fm[k]
    __shared__ float bs1[SDIM];
    __shared__ float bv[SDIM * 2];                     // b[j][q]

    const int t   = threadIdx.x;
    const int gid = blockIdx.x;
    const int i   = gid >> 3;     // 0..127
    const int bb  = gid & 7;      // 0..7

    // ---------------- init ----------------
    const int j = t >> 2;         // this thread's j row (elementwise work)
    const int r = t & 3;          // quarter-row, k = r*32 + n
    float* m0row = &sm0[j * MSTR + r * 32];
    float* m1row = &sm1[j * MSTR + r * 32];
    #pragma unroll
    for (int n = 0; n < 32; ++n) { m0row[n] = 0.0f; m1row[n] = 0.0f; }

    if (t < SDIM) {
        fm[t] = mask[(bb * SDIM + t) * SDIM + i] ? 1.0f : 0.0f;
    }
    if (t < 2 * SDIM) {
        int jj = t >> 1, q = t & 1;
        pe[t] = s_edge[((bb * SDIM + jj) * SDIM + i) * 2 + q];
    }

    // softplus(s_sib[b, j, i, k]) into registers (q=1 additive term)
    float sp[32];
    {
        const float* srow = &s_sib[((bb * SDIM + j) * SDIM + i) * SDIM + r * 32];
        #pragma unroll
        for (int n = 0; n < 32; ++n) {
            float s = srow[n];
            sp[n] = fmaxf(s, 0.0f) + log1pf(expf(-fabsf(s)));
        }
    }
    __syncthreads();

    // WMMA mapping: 16 waves -> 8 j-tiles x 2 q-channels
    const int lane = t & 31;
    const int wv   = t >> 5;
    const int jt   = wv & 7;               // j tile (16 rows)
    const int qw   = wv >> 3;              // q channel
    const int row  = lane & 15;            // A-matrix M row within tile
    const int koff = (lane >> 4) << 1;     // lanes 0-15: K0/K1, lanes 16-31: K2/K3
    const float* abase = (qw ? sm1 : sm0) + (jt * 16 + row) * MSTR;
    float* bsout = qw ? bs1 : bs0;

    for (int iter = 0; iter < 3; ++iter) {
        // ---- bs_q[j] = sum_k m_q[j][k] * fm[k] via V_WMMA_F32_16X16X4_F32 ----
        v8f c = {};
        for (int kk = 0; kk < 32; ++kk) {
            int kb = kk * 4 + koff;
            v2f a = *(const v2f*)(abase + kb);
            v2f bm;
            bm.x = fm[kb];       // B[k][c] = fm[k] for all 16 columns
            bm.y = fm[kb + 1];
            c = __builtin_amdgcn_wmma_f32_16x16x4_f32(
                    false, a, false, bm, (short)0, c, false, false);
        }
        // D column 0: lane 0 holds rows 0..7 (VGPR 0..7), lane 16 holds rows 8..15
        if (lane == 0 || lane == 16) {
            int rb = (lane >> 4) * 8;
            #pragma unroll
            for (int rr = 0; rr < 8; ++rr)
                bsout[jt * 16 + rb + rr] = c[rr];
        }
        __syncthreads();

        // ---- b[j][q] = fm[j] * (p_edge[j][q] + bs_q[j]) ----
        if (t < 2 * SDIM) {
            int jj = t >> 1, q = t & 1;
            float ssum = q ? bs1[jj] : bs0[jj];
            bv[t] = fm[jj] * (pe[t] + ssum);
        }
        __syncthreads();

        if (iter < 2) {  // final m-update is dead code in the reference
            // t0[j,k,q] = b[k,q] - m[j,k,q];  a0 = t0_0 + ln2; a1 = t0_1 + softplus(s)
            // m_new = a - logsumexp(a0, a1)
            const float* bvk = &bv[(r * 32) * 2];
            #pragma unroll
            for (int n = 0; n < 32; ++n) {
                float b0  = bvk[2 * n + 0];
                float b1  = bvk[2 * n + 1];
                float a0v = b0 - m0row[n] + LN2F;
                float a1v = b1 - m1row[n] + sp[n];
                float mx  = fmaxf(a0v, a1v);
                float lse = mx + log1pf(expf(-fabsf(a0v - a1v)));
                m0row[n] = a0v - lse;
                m1row[n] = a1v - lse;
            }
            __syncthreads();
        }
    }

    // ---- marginals[bb, j, i, q] = b[i, j, bb, q] ----
    if (t < 2 * SDIM) {
        int jj = t >> 1, q = t & 1;
        out[((bb * SDIM + jj) * SDIM + i) * 2 + q] = bv[t];
    }
}

extern "C" void kernel_launch(void* const* d_in, const int* in_sizes, int n_in,
                              void* d_out, int out_size, void* d_ws, size_t ws_size,
                              hipStream_t stream) {
    (void)in_sizes; (void)n_in; (void)out_size; (void)d_ws; (void)ws_size;
    const float*         s_edge = (const float*)d_in[0];
    const float*         s_sib  = (const float*)d_in[1];
    const unsigned char* mask   = (const unsigned char*)d_in[2];  // numpy bool = 1 byte
    float*               out    = (float*)d_out;

    // one block per (i, b) pair: 128 * 8 = 1024 blocks, 512 threads (16 waves)
    lbp_fused_kernel<<<dim3(SDIM * NBATCH), dim3(512), 0, stream>>>(
        s_edge, s_sib, mask, out);
}